// CTForwardModel_58299886076394
// MI455X (gfx1250) — compile-verified
//
#include <hip/hip_runtime.h>

#define IMG 256
#define NA  180
#define ND  363
#define PITCH 260                  // LDS row pitch in floats: 1040B rows, 16B-aligned for B128
#define ANG_PER_BLK 2

#if __has_builtin(__builtin_amdgcn_wmma_f32_16x16x4_f32)
#define HAVE_WMMA 1
#endif
#if __has_builtin(__builtin_amdgcn_global_load_async_to_lds_b128) && __has_builtin(__builtin_amdgcn_s_wait_asynccnt)
#define HAVE_ASYNC 1
#endif

typedef __attribute__((ext_vector_type(2))) float v2f;
typedef __attribute__((ext_vector_type(8))) float v8f;
typedef __attribute__((ext_vector_type(4))) int   i4;
typedef __attribute__((address_space(1))) i4*     g4p;   // int4 in global memory
typedef __attribute__((address_space(3))) i4*     l4p;   // int4 in LDS

__device__ __forceinline__ float tap(const float* simg, int y, int x) {
  bool ok = ((unsigned)y < (unsigned)IMG) & ((unsigned)x < (unsigned)IMG);
  int yc = min(max(y, 0), IMG - 1);
  int xc = min(max(x, 0), IMG - 1);
  float v = simg[yc * PITCH + xc];
  return ok ? v : 0.0f;
}

// Bilinear sample of the LDS-staged image at the rotated coordinate for
// meshgrid row i, column coordinate xj, rotation (ct, st) = (cos(-ang), sin(-ang)).
__device__ __forceinline__ float bsample(const float* simg, int i, float xj, float ct, float st) {
  float yi = -1.0f + (2.0f / 255.0f) * (float)i;
  float gx = xj * ct - yi * st;
  float gy = xj * st + yi * ct;
  float fx = (gx + 1.0f) * 127.5f;
  float fy = (gy + 1.0f) * 127.5f;
  float f0x = floorf(fx), f0y = floorf(fy);
  int   x0 = (int)f0x,  y0 = (int)f0y;
  float wx = fx - f0x,  wy = fy - f0y;
  float v00 = tap(simg, y0,     x0);
  float v01 = tap(simg, y0,     x0 + 1);
  float v10 = tap(simg, y0 + 1, x0);
  float v11 = tap(simg, y0 + 1, x0 + 1);
  return v00 * (1.0f - wy) * (1.0f - wx)
       + v01 * (1.0f - wy) * wx
       + v10 * wy * (1.0f - wx)
       + v11 * wy * wx;
}

__global__ __launch_bounds__(256, 1)
void radon_kernel(const float* __restrict__ x, float* __restrict__ out) {
  __shared__ float simg[IMG * PITCH];   // 266,240 B: whole image, pitch-padded (fits 320KB LDS)
  __shared__ float proj[IMG];           // one projection row (256 floats)

  const int b   = blockIdx.x / (NA / ANG_PER_BLK);
  const int ab  = blockIdx.x % (NA / ANG_PER_BLK);
  const int tid = threadIdx.x;
  const float* img = x + (size_t)b * IMG * IMG;

  // ---- Stage the full 256KB image into LDS (async DMA path on CDNA5) ----
  for (int c = tid; c < IMG * (IMG / 4); c += 256) {   // 16384 x 16B chunks, 64 per thread
    int row = c >> 6;
    int q   = (c & 63) << 2;                           // starting float within row
#ifdef HAVE_ASYNC
    __builtin_amdgcn_global_load_async_to_lds_b128(
        (g4p)(img + row * IMG + q),
        (l4p)(&simg[row * PITCH + q]),
        0, 0);
#else
    *(float4*)(&simg[row * PITCH + q]) = *(const float4*)(img + row * IMG + q);
#endif
  }
#ifdef HAVE_ASYNC
  __builtin_amdgcn_s_wait_asynccnt(0);
#endif
  __syncthreads();

  const int lane = tid & 31;
  const int wv   = tid >> 5;
  const int half = lane >> 4;   // which 16-lane half of the wave
  const int col  = lane & 15;   // column within a 16-wide j-block

  for (int al = 0; al < ANG_PER_BLK; ++al) {
    const int   a   = ab * ANG_PER_BLK + al;
    const float ang = -(float)a * 0.017453292519943295f;  // -angle, matches cos(-t)/sin(-t)
    const float ct  = cosf(ang);
    const float st  = sinf(ang);

#ifdef HAVE_WMMA
    // proj[j] = sum_i sampled[i][j]  ==  ones(16x4) x B(4x16) accumulated over 64 K-chunks.
    // With A == ones, D(m,n) = sum_k B(k,n): result is independent of which K-slot a lane
    // holds, as long as lanes {n, n+16} jointly supply the 4 row-samples of column n.
    v2f ones; ones.x = 1.0f; ones.y = 1.0f;
    for (int t = 0; t < 2; ++t) {                 // 8 waves x 2 j-blocks = 256 columns
      const int   jb = (wv * 2 + t) * 16;
      const float xj = -1.0f + (2.0f / 255.0f) * (float)(jb + col);
      v8f c = {};
      for (int k = 0; k < 64; ++k) {
        const int i0 = 4 * k + 2 * half;          // lanes 0-15: rows 4k,4k+1; lanes 16-31: 4k+2,4k+3
        v2f bv;
        bv.x = bsample(simg, i0,     xj, ct, st);
        bv.y = bsample(simg, i0 + 1, xj, ct, st);
        c = __builtin_amdgcn_wmma_f32_16x16x4_f32(false, ones, false, bv,
                                                  (short)0, c, false, false);
      }
      // Every row of D holds the column sum; lanes 0-15 carry columns N=0..15 in c[0].
      if (lane < 16) proj[jb + lane] = c[0];
    }
#else
    {
      const float xj = -1.0f + (2.0f / 255.0f) * (float)tid;
      float s = 0.0f;
      for (int i = 0; i < IMG; ++i) s += bsample(simg, i, xj, ct, st);
      proj[tid] = s;
    }
#endif
    __syncthreads();

    // ---- Detector resampling: sino[d] = lerp(proj[p0], proj[p1], w) ----
    for (int d = tid; d < ND; d += 256) {
      float pos = (float)d * (255.0f / 362.0f);
      int   p0  = (int)pos;
      float w   = pos - (float)p0;
      int   p1  = min(p0 + 1, IMG - 1);
      out[(size_t)(b * NA + a) * ND + d] = proj[p0] * (1.0f - w) + proj[p1] * w;
    }
    __syncthreads();
  }
}

extern "C" void kernel_launch(void* const* d_in, const int* in_sizes, int n_in,
                              void* d_out, int out_size, void* d_ws, size_t ws_size,
                              hipStream_t stream) {
  (void)in_sizes; (void)n_in; (void)out_size; (void)d_ws; (void)ws_size;
  const float* x = (const float*)d_in[0];
  float* out = (float*)d_out;
  const int blocks = 4 * (NA / ANG_PER_BLK);   // 360 workgroups of 256 threads (8 waves)
  radon_kernel<<<dim3(blocks), dim3(256), 0, stream>>>(x, out);
}